// CalibratedNorm_76192719831876
// MI455X (gfx1250) — compile-verified
//
#include <hip/hip_runtime.h>
#include <hip/hip_bf16.h>
#include <math.h>

// Shapes from reference
#define B_   32
#define C_   256
#define H_   56
#define W_   56
#define HW_  (H_ * W_)          // 3136
#define HW4_ (HW_ / 4)          // 784 float4 per plane
#define NPLANE_ (B_ * C_)       // 8192
#define EPS_ 1e-05f

typedef float  f4  __attribute__((ext_vector_type(4)));
typedef float  v2f __attribute__((ext_vector_type(2)));
typedef float  v8f __attribute__((ext_vector_type(8)));

// ---------------------------------------------------------------------------
// Kernel 1: per-(b,c) spatial sum. One block per plane. b128 loads, LDS tree.
// Reads all of x (103 MB) -> leaves x hot in the 192MB L2 for k_apply.
// ---------------------------------------------------------------------------
__global__ void cn_reduce_kernel(const float* __restrict__ x,
                                 float* __restrict__ sums) {
    const int plane = blockIdx.x;                 // b*C + c
    const int tid   = threadIdx.x;                // 0..255
    const f4* __restrict__ xp =
        (const f4*)(x + (size_t)plane * HW_);     // 16B aligned (3136*4 % 16 == 0)

    float s = 0.0f;
    for (int i = tid; i < HW4_; i += 256) {
        f4 v = xp[i];
        s += v.x + v.y + v.z + v.w;
    }

    __shared__ float red[256];
    red[tid] = s;
    __syncthreads();
    for (int off = 128; off > 0; off >>= 1) {
        if (tid < off) red[tid] += red[tid + off];
        __syncthreads();
    }
    if (tid == 0) sums[plane] = red[0];
}

// ---------------------------------------------------------------------------
// Kernel 2: per-channel scale/shift tables.
//   gs  = g_w * rsqrt(g_rv + eps)        gsh = g_b - g_rm * gs
//   ms  = mean_g(grp_w * rsqrt(grp_rv))  msh = mean_g(grp_b - grp_rm * sg)
// ---------------------------------------------------------------------------
__global__ void cn_tables_kernel(const float* __restrict__ g_w,
                                 const float* __restrict__ g_b,
                                 const float* __restrict__ g_rm,
                                 const float* __restrict__ g_rv,
                                 const float* __restrict__ grp_w,
                                 const float* __restrict__ grp_b,
                                 const float* __restrict__ grp_rm,
                                 const float* __restrict__ grp_rv,
                                 float* __restrict__ gs,
                                 float* __restrict__ gsh,
                                 float* __restrict__ ms,
                                 float* __restrict__ msh) {
    const int c = threadIdx.x;                    // 256 threads, 1 block
    const float s = g_w[c] * rsqrtf(g_rv[c] + EPS_);
    gs[c]  = s;
    gsh[c] = g_b[c] - g_rm[c] * s;

    float sm = 0.0f, smh = 0.0f;
    for (int g = 0; g < 32; ++g) {
        const int idx = g * C_ + c;
        const float sg = grp_w[idx] * rsqrtf(grp_rv[idx] + EPS_);
        sm  += sg;
        smh += grp_b[idx] - grp_rm[idx] * sg;
    }
    ms[c]  = sm  * (1.0f / 32.0f);
    msh[c] = smh * (1.0f / 32.0f);
}

// ---------------------------------------------------------------------------
// Kernel 3: alpha[b] = sigmoid(pooled(32x256) @ alpha_w + alpha_b) via
// V_WMMA_F32_16X16X4_F32.  One full wave (32 lanes), EXEC all ones.
//
// Layout-robust formulation:  A[m,k] = sums[m, c0+k] * (alpha_w[c0+k]/HW),
// B = all ones (4x16) -> D[m,n] = sum_k A[m,k], independent of B's layout.
// A 16x4 f32 layout (ISA 7.12.2): lanes 0-15 hold K=0,1 in VGPR0,1 for row
// M=lane; lanes 16-31 hold K=2,3 for row M=lane-16.
// D 16x16 f32 layout: VGPR r = row r (lanes 0-15, col=lane) / row r+8
// (lanes 16-31). Column 0 lives in lanes 0 and 16.
// Two accumulators: acc0 -> b=0..15, acc1 -> b=16..31.
// ---------------------------------------------------------------------------
__global__ void cn_alpha_wmma_kernel(const float* __restrict__ sums,
                                     const float* __restrict__ alpha_w,
                                     const float* __restrict__ alpha_b,
                                     float* __restrict__ alpha_out) {
    const int lane = threadIdx.x;                 // 0..31, one wave32
    const int m0   = lane & 15;                   // A row for this lane
    const int kb   = (lane >> 4) << 1;            // K sub-offset: 0 or 2
    const float inv = 1.0f / (float)HW_;

    v8f acc0 = {};
    v8f acc1 = {};
    const v2f bones = {1.0f, 1.0f};

    for (int c0 = 0; c0 < C_; c0 += 4) {
        const float w0 = alpha_w[c0 + kb]     * inv;
        const float w1 = alpha_w[c0 + kb + 1] * inv;

        v2f a0, a1;
        a0.x = sums[(size_t)m0 * C_        + c0 + kb]     * w0;
        a0.y = sums[(size_t)m0 * C_        + c0 + kb + 1] * w1;
        a1.x = sums[(size_t)(m0 + 16) * C_ + c0 + kb]     * w0;
        a1.y = sums[(size_t)(m0 + 16) * C_ + c0 + kb + 1] * w1;

        acc0 = __builtin_amdgcn_wmma_f32_16x16x4_f32(
            false, a0, false, bones, (short)0, acc0, false, false);
        acc1 = __builtin_amdgcn_wmma_f32_16x16x4_f32(
            false, a1, false, bones, (short)0, acc1, false, false);
    }

    // Lanes 0 and 16 hold column 0 of D: VGPR r -> row r (lane 0) / r+8 (lane 16).
    if ((lane & 15) == 0) {
        const int base = (lane == 0) ? 0 : 8;
        const float ab = alpha_b[0];
        #pragma unroll
        for (int r = 0; r < 8; ++r) {
            const float z0 = acc0[r] + ab;        // batch base + r
            const float z1 = acc1[r] + ab;        // batch 16 + base + r
            alpha_out[base + r]      = 1.0f / (1.0f + __expf(-z0));
            alpha_out[16 + base + r] = 1.0f / (1.0f + __expf(-z1));
        }
    }
}

// ---------------------------------------------------------------------------
// Kernel 4: out = x * lerp(gs,ms,alpha) + lerp(gsh,msh,alpha), per plane.
// x reads should hit L2 (loaded by cn_reduce); output written with
// non-temporal b128 stores so the 103MB of output doesn't evict x from L2.
// ---------------------------------------------------------------------------
__global__ void cn_apply_kernel(const float* __restrict__ x,
                                const float* __restrict__ alpha,
                                const float* __restrict__ gs,
                                const float* __restrict__ gsh,
                                const float* __restrict__ ms,
                                const float* __restrict__ msh,
                                float* __restrict__ out) {
    const int plane = blockIdx.x;                 // b*C + c
    const int b = plane >> 8;
    const int c = plane & (C_ - 1);
    const int tid = threadIdx.x;

    const float a  = alpha[b];
    const float s  = (1.0f - a) * gs[c]  + a * ms[c];
    const float sh = (1.0f - a) * gsh[c] + a * msh[c];

    const f4* __restrict__ xp = (const f4*)(x   + (size_t)plane * HW_);
    f4* __restrict__       op = (f4*)      (out + (size_t)plane * HW_);

    for (int i = tid; i < HW4_; i += 256) {
        f4 v = xp[i];
        f4 r;
        r.x = v.x * s + sh;
        r.y = v.y * s + sh;
        r.z = v.z * s + sh;
        r.w = v.w * s + sh;
        __builtin_nontemporal_store(r, op + i);
    }
}

// ---------------------------------------------------------------------------
// Launch. Inputs (setup_inputs order):
//  0:x 1:labels 2:alpha_w 3:alpha_b 4:g_w 5:g_b 6:g_rm 7:g_rv
//  8:grp_w 9:grp_b 10:grp_rm 11:grp_rv
// ws layout (floats): sums[8192] | alpha[32] | gs[256] | gsh[256] | ms[256] | msh[256]
// = 9248 floats (~37 KB) — well under any reasonable ws_size.
// ---------------------------------------------------------------------------
extern "C" void kernel_launch(void* const* d_in, const int* in_sizes, int n_in,
                              void* d_out, int out_size, void* d_ws, size_t ws_size,
                              hipStream_t stream) {
    const float* x       = (const float*)d_in[0];
    const float* alpha_w = (const float*)d_in[2];
    const float* alpha_b = (const float*)d_in[3];
    const float* g_w     = (const float*)d_in[4];
    const float* g_b     = (const float*)d_in[5];
    const float* g_rm    = (const float*)d_in[6];
    const float* g_rv    = (const float*)d_in[7];
    const float* grp_w   = (const float*)d_in[8];
    const float* grp_b   = (const float*)d_in[9];
    const float* grp_rm  = (const float*)d_in[10];
    const float* grp_rv  = (const float*)d_in[11];
    float* out = (float*)d_out;

    float* ws    = (float*)d_ws;
    float* sums  = ws;                  // 8192
    float* alpha = ws + NPLANE_;        // 32
    float* gs    = alpha + 32;          // 256
    float* gsh   = gs + C_;             // 256
    float* ms    = gsh + C_;            // 256
    float* msh   = ms + C_;             // 256

    cn_reduce_kernel<<<NPLANE_, 256, 0, stream>>>(x, sums);
    cn_tables_kernel<<<1, 256, 0, stream>>>(g_w, g_b, g_rm, g_rv,
                                            grp_w, grp_b, grp_rm, grp_rv,
                                            gs, gsh, ms, msh);
    cn_alpha_wmma_kernel<<<1, 32, 0, stream>>>(sums, alpha_w, alpha_b, alpha);
    cn_apply_kernel<<<NPLANE_, 256, 0, stream>>>(x, alpha, gs, gsh, ms, msh, out);
}